// PanopticHead_12429635355107
// MI455X (gfx1250) — compile-verified
//
#include <hip/hip_runtime.h>

#define HH 256
#define WW 512
#define NN 64
#define MM 100
#define CTH 80
#define CST 53
#define HWQ (HH * WW / 4)

#define MP 112   // mask rows / GEMM2 K, padded to 7*16
#define BP 80    // padded box extent (max box side is 76), 5*16

typedef __attribute__((ext_vector_type(2))) float v2f;
typedef __attribute__((ext_vector_type(8))) float v8f;

// ---------------------------------------------------------------------------
// Kernel 1: base image. Channels 0..52: copy stuff logits. Channels 53..116
// (instance n): cropped thing-sem channel inside the (round-half-even) crop
// box, zero elsewhere. Streaming float4 — this kernel carries ~all the HBM
// bandwidth (~120 MB total traffic -> ~5.4 us at 23.3 TB/s).
// ---------------------------------------------------------------------------
__global__ void pan_base_kernel(const float* __restrict__ sem,
                                const float* __restrict__ bbox,
                                const int* __restrict__ cls_idx,
                                float* __restrict__ out) {
  long idx = (long)blockIdx.x * blockDim.x + threadIdx.x;
  const long total = (long)(CST + NN) * HWQ;
  if (idx >= total) return;
  int c = (int)(idx / HWQ);
  int rem = (int)(idx % HWQ);
  const float4* sem4 = (const float4*)sem;
  float4* out4 = (float4*)out;
  if (c < CST) {
    out4[idx] = sem4[(long)c * HWQ + rem];
    return;
  }
  int n = c - CST;
  int y = rem / (WW / 4);
  int x4 = (rem % (WW / 4)) * 4;
  float b0 = bbox[n * 4 + 0], b1 = bbox[n * 4 + 1];
  float b2 = bbox[n * 4 + 2], b3 = bbox[n * 4 + 3];
  int x1 = (int)b0;                 // positive -> trunc == floor (matches astype)
  int y1 = (int)b1;
  int x2 = (int)rintf(b2) + 1;      // rintf: round-half-even, matches jnp.round
  int y2 = (int)rintf(b3) + 1;
  int cls = cls_idx[n];
  const float* thing = sem + (long)(CST + cls) * (HH * WW) + (long)y * WW;
  bool rowin = (y >= y1) && (y < y2);
  float4 v;
  float* vp = (float*)&v;
#pragma unroll
  for (int j = 0; j < 4; ++j) {
    int xx = x4 + j;
    vp[j] = (rowin && xx >= x1 && xx < x2) ? thing[xx] : 0.0f;
  }
  out4[idx] = v;
}

// ---------------------------------------------------------------------------
// Kernel 2: bilinear paste as two fp32 WMMA GEMMs per instance.
//   Out(Hbox x Wbox) = Wy(Hbox x 100) * mask(100 x 100) * Wx^T(100 x Wbox)
// Wy/Wx rows hold the two _idx_frac lerp taps; padded rows/cols stay zero so
// padded tiles contribute nothing. One workgroup (8 wave32) per instance.
// Adds into the base image written by kernel 1 (same stream => ordered).
// ---------------------------------------------------------------------------
__global__ void pan_paste_wmma(const float* __restrict__ mask_logits,
                               const float* __restrict__ bbox,
                               const int* __restrict__ cls_idx,
                               float* __restrict__ out) {
  extern __shared__ float sm[];
  float* maskS = sm;                  // 112 x 100 (rows 100..111 zeroed)
  float* wxT   = maskS + MP * MM;     // 100 x 80  (row j, col x)
  float* wy    = wxT + MM * BP;       // 80 x 112  (row y, col i)
  float* Tm    = wy + BP * MP;        // 112 x 80  intermediate mask*Wx^T

  const int n = blockIdx.x;
  const int tid = threadIdx.x;

  float b0 = bbox[n * 4 + 0], b1 = bbox[n * 4 + 1];
  float b2 = bbox[n * 4 + 2], b3 = bbox[n * 4 + 3];
  int x1b = (int)floorf(b0), y1b = (int)floorf(b1);
  int x2b = (int)floorf(b2), y2b = (int)floorf(b3);
  int Wbox = x2b - x1b + 1;
  int Hbox = y2b - y1b + 1;
  float sX = (float)MM / (float)Wbox;
  float sY = (float)MM / (float)Hbox;
  int cls = cls_idx[n];
  const float* msrc = mask_logits + ((long)n * CTH + cls) * (MM * MM);

  // Stage mask into LDS (zero the 12 pad rows); zero the weight matrices.
  for (int i = tid; i < MP * MM; i += blockDim.x)
    maskS[i] = (i < MM * MM) ? msrc[i] : 0.0f;
  for (int i = tid; i < MM * BP; i += blockDim.x) wxT[i] = 0.0f;
  for (int i = tid; i < BP * MP; i += blockDim.x) wy[i] = 0.0f;
  __syncthreads();

  // Scatter the 2-tap interpolation weights (exact _idx_frac semantics).
  if (tid < Wbox && tid < BP) {
    float mx = ((float)tid + 0.5f) * sX - 0.5f;
    mx = fmaxf(mx, 0.0f);
    float c0 = floorf(mx);
    float fx = mx - c0;
    int i0 = min((int)c0, MM - 1);
    int i1 = min((int)c0 + 1, MM - 1);
    wxT[i0 * BP + tid] += 1.0f - fx;
    wxT[i1 * BP + tid] += fx;
  }
  if (tid < Hbox && tid < BP) {
    float my = ((float)tid + 0.5f) * sY - 0.5f;
    my = fmaxf(my, 0.0f);
    float c0 = floorf(my);
    float fy = my - c0;
    int i0 = min((int)c0, MM - 1);
    int i1 = min((int)c0 + 1, MM - 1);
    wy[tid * MP + i0] += 1.0f - fy;
    wy[tid * MP + i1] += fy;
  }
  __syncthreads();

  const int lane = tid & 31;
  const int wv = tid >> 5;            // wave id 0..7 (tiles assigned per wave,
  const int row = lane & 15;          //  keeps EXEC all-ones around WMMA)
  const int hi = lane >> 4;           // half-wave select
  const int koff = hi << 1;           // A/B K offset: lanes 0-15 -> K,K+1; 16-31 -> K+2,K+3

  // GEMM1: Tm(112x80) = maskS(112x100) x wxT(100x80), K = 100 = 25 * 4.
  for (int t = wv; t < 7 * 5; t += 8) {
    int i0 = (t / 5) * 16;
    int x0 = (t % 5) * 16;
    v8f acc = {};
    for (int k = 0; k < MM; k += 4) {
      int kk = k + koff;
      v2f A, B;
      A[0] = maskS[(i0 + row) * MM + kk];
      A[1] = maskS[(i0 + row) * MM + kk + 1];
      B[0] = wxT[kk * BP + x0 + row];
      B[1] = wxT[(kk + 1) * BP + x0 + row];
      acc = __builtin_amdgcn_wmma_f32_16x16x4_f32(false, A, false, B,
                                                  (short)0, acc, false, false);
    }
#pragma unroll
    for (int r = 0; r < 8; ++r)
      Tm[(i0 + r + (hi << 3)) * BP + x0 + row] = acc[r];
  }
  __syncthreads();

  // GEMM2: Out(80x80) = wy(80x112) x Tm(112x80), K = 112 = 28 * 4.
  for (int t = wv; t < 5 * 5; t += 8) {
    int y0 = (t / 5) * 16;
    int x0 = (t % 5) * 16;
    v8f acc = {};
    for (int k = 0; k < MP; k += 4) {
      int kk = k + koff;
      v2f A, B;
      A[0] = wy[(y0 + row) * MP + kk];
      A[1] = wy[(y0 + row) * MP + kk + 1];
      B[0] = Tm[kk * BP + x0 + row];
      B[1] = Tm[(kk + 1) * BP + x0 + row];
      acc = __builtin_amdgcn_wmma_f32_16x16x4_f32(false, A, false, B,
                                                  (short)0, acc, false, false);
    }
#pragma unroll
    for (int r = 0; r < 8; ++r) {
      int yl = y0 + r + (hi << 3);
      int xl = x0 + row;
      if (yl < Hbox && xl < Wbox) {
        long o = ((long)(CST + n) * HH + (y1b + yl)) * WW + (x1b + xl);
        out[o] += acc[r];
      }
    }
  }
}

extern "C" void kernel_launch(void* const* d_in, const int* in_sizes, int n_in,
                              void* d_out, int out_size, void* d_ws, size_t ws_size,
                              hipStream_t stream) {
  const float* mask_logits = (const float*)d_in[0];   // (64,80,100,100)
  const float* sem         = (const float*)d_in[1];   // (1,133,256,512)
  const float* bbox        = (const float*)d_in[2];   // (64,4)
  const int*   cls_idx     = (const int*)d_in[3];     // (64,)
  float* out = (float*)d_out;                          // (1,117,256,512)

  (void)in_sizes; (void)n_in; (void)out_size; (void)d_ws; (void)ws_size;

  // Base image: 117 channels * 32768 float4 lanes.
  long total = (long)(CST + NN) * HWQ;
  int blocks = (int)((total + 255) / 256);
  pan_base_kernel<<<blocks, 256, 0, stream>>>(sem, bbox, cls_idx, out);

  // Paste: one workgroup per instance, ~145 KB dynamic LDS (WGP has 320 KB).
  size_t smem = (size_t)(MP * MM + MM * BP + BP * MP + MP * BP) * sizeof(float);
  hipFuncSetAttribute((const void*)pan_paste_wmma,
                      hipFuncAttributeMaxDynamicSharedMemorySize, (int)smem);
  pan_paste_wmma<<<NN, 256, smem, stream>>>(mask_logits, bbox, cls_idx, out);
}